// DeformRoiPoolExtractor_1202590843766
// MI455X (gfx1250) — compile-verified
//
#include <hip/hip_runtime.h>

typedef __attribute__((ext_vector_type(2))) float v2f;
typedef __attribute__((ext_vector_type(8))) float v8f;

#define ROI_OUT   7
#define NBINS     49     // 7*7
#define NBINS_PAD 64
#define NCH       256
#define KTAPS     1024   // 64 padded bins * 16 taps

// One workgroup per ROI: 256 threads = 8 wave32.
// Phase 1: build 1024 (weight, pixel-offset) tap pairs in LDS.
// Phase 2: out[bin,c] = sum_p W[bin,p] * F[p,c] via V_WMMA_F32_16X16X4_F32,
//          W block-diagonal (16 taps per bin), K streamed 4 taps at a time.
__global__ __launch_bounds__(256) void deform_roi_pool_wmma(
    const float* __restrict__ f0, const float* __restrict__ f1,
    const float* __restrict__ f2, const float* __restrict__ f3,
    const float* __restrict__ rois, float* __restrict__ out)
{
    __shared__ float sW[KTAPS];
    __shared__ int   sOff[KTAPS];

    const int roi = blockIdx.x;
    const int tid = threadIdx.x;

    // ---- per-ROI params (redundant per thread; cheap) ----
    const float bidxf = rois[roi * 5 + 0];
    const float rx1   = rois[roi * 5 + 1];
    const float ry1   = rois[roi * 5 + 2];
    const float rx2   = rois[roi * 5 + 3];
    const float ry2   = rois[roi * 5 + 4];

    const float scale = sqrtf((rx2 - rx1) * (ry2 - ry1));
    const float tlf   = floorf(log2f(scale * (1.0f / 56.0f) + 1e-6f));
    const int   lvl   = (int)fminf(fmaxf(tlf, 0.0f), 3.0f);

    const int   H  = 200 >> lvl;           // 200,100,50,25
    const int   W  = H;
    const int   HW = H * W;
    const float sc = 1.0f / (float)(4 << lvl);  // 1/4,1/8,1/16,1/32

    const float* feat = (lvl == 0) ? f0 : (lvl == 1) ? f1 : (lvl == 2) ? f2 : f3;
    const float* base = feat + (long)((int)bidxf) * NCH * HW;  // uniform per block

    const float x1 = rx1 * sc - 0.5f;
    const float y1 = ry1 * sc - 0.5f;
    const float bw = (rx2 * sc - 0.5f - x1) * (1.0f / ROI_OUT);
    const float bh = (ry2 * sc - 0.5f - y1) * (1.0f / ROI_OUT);

    // ---- phase 1: tap table (4 taps per thread) ----
    for (int p = tid; p < KTAPS; p += 256) {
        const int b = p >> 4;              // padded bin id
        float wgt = 0.0f;
        int   off = 0;
        if (b < NBINS) {
            const int s  = (p >> 2) & 3;   // sample in 2x2
            const int t  = p & 3;          // bilinear tap
            const int by = b / ROI_OUT, bx = b % ROI_OUT;
            const float gy = (float)by + ((float)(s >> 1) + 0.5f) * 0.5f;
            const float gx = (float)bx + ((float)(s & 1) + 0.5f) * 0.5f;
            const float y  = y1 + bh * gy;
            const float x  = x1 + bw * gx;
            const bool valid = (y > -1.0f) && (y < (float)H) &&
                               (x > -1.0f) && (x < (float)W);
            const float yc = fminf(fmaxf(y, 0.0f), (float)(H - 1));
            const float xc = fminf(fmaxf(x, 0.0f), (float)(W - 1));
            const int y0  = min((int)floorf(yc), H - 1);
            const int x0  = min((int)floorf(xc), W - 1);
            const int y1i = min(y0 + 1, H - 1);
            const int x1i = min(x0 + 1, W - 1);
            const float ly = yc - (float)y0, lx = xc - (float)x0;
            const float wy = (t & 2) ? ly : (1.0f - ly);
            const float wx = (t & 1) ? lx : (1.0f - lx);
            const int   yy = (t & 2) ? y1i : y0;
            const int   xx = (t & 1) ? x1i : x0;
            wgt = valid ? (wy * wx * 0.25f) : 0.0f;   // 0.25 = mean over 4 samples
            off = yy * W + xx;
        }
        sW[p]   = wgt;
        sOff[p] = off;
    }
    __syncthreads();

    // ---- phase 2: block-diagonal GEMM on the matrix pipe ----
    const int wave = tid >> 5;
    const int lane = tid & 31;
    const int half = lane >> 4;   // 0: K rows 0,1   1: K rows 2,3
    const int n    = lane & 15;   // A row (bin-in-tile) == B col (ch-in-tile)

    for (int tile = wave; tile < NBINS_PAD; tile += 8) {
        const int m  = tile >> 4;           // bin tile 0..3
        const int ct = tile & 15;           // channel tile 0..15
        const int c  = ct * 16 + n;
        const int cHW = c * HW;             // 32-bit offset from uniform base
        const int kbase = m * 256;          // this tile's 256 taps

        v8f acc = {};
        #pragma unroll 4
        for (int k = 0; k < 64; ++k) {
            const int r = k >> 2;           // only nonzero A row this K-step
            const int p = kbase + k * 4 + half * 2;
            const float w0 = sW[p];
            const float w1 = sW[p + 1];
            v2f a;                          // A 16x4 f32 operand
            a.x = (n == r) ? w0 : 0.0f;
            a.y = (n == r) ? w1 : 0.0f;
            v2f bmat;                       // B 4x16 f32 operand (gathered)
            bmat.x = base[cHW + sOff[p]];
            bmat.y = base[cHW + sOff[p + 1]];
            acc = __builtin_amdgcn_wmma_f32_16x16x4_f32(
                /*neg_a=*/false, a, /*neg_b=*/false, bmat,
                /*c_mod=*/(short)0, acc, /*reuse_a=*/false, /*reuse_b=*/false);
        }

        // D layout: VGPR i -> row (i + 8*half), col n.
        // Tiles m=0..2: all bins valid -> unconditional contiguous stores.
        // Tile m=3: only bin 48 (half==0, i==0) is in range.
        float* orow = out + ((long)roi * NCH + c) * NBINS + (m * 16 + half * 8);
        if (m < 3) {
            #pragma unroll
            for (int i = 0; i < 8; ++i) orow[i] = acc[i];
        } else if (half == 0) {
            orow[0] = acc[0];               // bin 48
        }
    }
}

extern "C" void kernel_launch(void* const* d_in, const int* in_sizes, int n_in,
                              void* d_out, int out_size, void* d_ws, size_t ws_size,
                              hipStream_t stream) {
    const float* f0   = (const float*)d_in[0];
    const float* f1   = (const float*)d_in[1];
    const float* f2   = (const float*)d_in[2];
    const float* f3   = (const float*)d_in[3];
    const float* rois = (const float*)d_in[4];
    float* out = (float*)d_out;
    const int nrois = in_sizes[4] / 5;
    deform_roi_pool_wmma<<<nrois, 256, 0, stream>>>(f0, f1, f2, f3, rois, out);
}